// ConvolutionLayers_88983132439254
// MI455X (gfx1250) — compile-verified
//
#include <hip/hip_runtime.h>
#include <hip/hip_bf16.h>

typedef float v2f __attribute__((ext_vector_type(2)));
typedef float v8f __attribute__((ext_vector_type(8)));

#define GCN_N 100000
#define GCN_E 1600000

// ---------------------------------------------------------------------------
// Degree / normalization
// ---------------------------------------------------------------------------
__global__ void k_init_deg(float* __restrict__ deg, int n) {
  int i = blockIdx.x * blockDim.x + threadIdx.x;
  if (i < n) deg[i] = 1.0f;  // self-loop contributes 1
}

__global__ void k_accum_deg(const int* __restrict__ dst, float* __restrict__ deg, int e) {
  int i = blockIdx.x * blockDim.x + threadIdx.x;
  if (i < e) unsafeAtomicAdd(&deg[dst[i]], 1.0f);  // hw global_atomic_add_f32
}

__global__ void k_dinv(float* __restrict__ deg, int n) {
  int i = blockIdx.x * blockDim.x + threadIdx.x;
  if (i < n) deg[i] = rsqrtf(deg[i]);  // deg >= 1 always (self loops)
}

// enorm[e] = dinv[src]*dinv[dst]  (precomputed once, reused by all 3 layers)
__global__ void k_edge_norm(const int* __restrict__ src, const int* __restrict__ dst,
                            const float* __restrict__ dinv, float* __restrict__ enorm, int e) {
  int i = blockIdx.x * blockDim.x + threadIdx.x;
  if (i < e) enorm[i] = dinv[src[i]] * dinv[dst[i]];
}

// ---------------------------------------------------------------------------
// Layer-1 transform (K=2 -> 64): pure VALU, 2 FMAs per output.
// One thread per (node, 4 channels).
// ---------------------------------------------------------------------------
__global__ void k_layer1(const float* __restrict__ x, const float* __restrict__ W1,
                         float* __restrict__ t, int n) {
  int tid = blockIdx.x * blockDim.x + threadIdx.x;
  if (tid >= n * 16) return;
  int i = tid >> 4;
  int c = (tid & 15) * 4;
  const float x0 = x[(size_t)i * 2];
  const float x1 = x[(size_t)i * 2 + 1];
  const float4 w0 = *(const float4*)(W1 + c);        // W1[0][c..c+3]
  const float4 w1 = *(const float4*)(W1 + 64 + c);   // W1[1][c..c+3]
  float4 o;
  o.x = x0 * w0.x + x1 * w1.x;
  o.y = x0 * w0.y + x1 * w1.y;
  o.z = x0 * w0.z + x1 * w1.z;
  o.w = x0 * w0.w + x1 * w1.w;
  *(float4*)(t + (size_t)i * 64 + c) = o;
}

// ---------------------------------------------------------------------------
// WMMA GEMM:  out[M x C] = (RELU_IN? relu(A) : A)[M x K] @ W[K x C] (+bias)(relu)
// K % 4 == 0, C % 16 == 0, rows covered exactly (M % 16 == 0).
// One wave computes one 16x16 tile with K/4 V_WMMA_F32_16X16X4_F32 ops.
// K and C are compile-time so all global loads use immediate offsets.
// VGPR layouts per CDNA5 ISA 7.12.2:
//   A 16x4 : g=lane>>4, m=lane&15 ; reg j holds A[m][2g+j]
//   B 4x16 : g, n=lane&15         ; reg j holds B[2g+j][n]
//   C 16x16: reg r -> (M=r+8g, N=lane&15)
// ---------------------------------------------------------------------------
template<int K, int C, bool RELU_IN, bool RELU_OUT, bool HAS_BIAS>
__global__ __launch_bounds__(128) void k_gemm_wmma(
    const float* __restrict__ A, const float* __restrict__ W,
    const float* __restrict__ bias, float* __restrict__ out) {
  const int lane  = threadIdx.x;                 // 0..31 (wave32)
  const int ntile = blockIdx.y * 4 + threadIdx.y;
  if (ntile >= (C >> 4)) return;                 // wave-uniform exit

  const int row0 = blockIdx.x * 16;
  const int g    = lane >> 4;
  const int mn   = lane & 15;
  const int col0 = ntile * 16;

  const float* __restrict__ Arow = A + (size_t)(row0 + mn) * K + 2 * g;  // lane's A row slice
  const float* __restrict__ Wcol = W + (size_t)(2 * g) * C + col0 + mn;  // lane's B col slice

  v8f acc = {};
  #pragma unroll
  for (int kb = 0; kb < K; kb += 4) {
    v2f av = *(const v2f*)(Arow + kb);            // A[m][kb+2g], A[m][kb+2g+1]
    if (RELU_IN) { av.x = fmaxf(av.x, 0.f); av.y = fmaxf(av.y, 0.f); }
    v2f bv;
    bv.x = Wcol[(size_t)kb * C];                  // B[kb+2g][n]
    bv.y = Wcol[(size_t)(kb + 1) * C];            // B[kb+2g+1][n]
    acc = __builtin_amdgcn_wmma_f32_16x16x4_f32(
        /*neg_a=*/false, av, /*neg_b=*/false, bv,
        /*c_mod=*/(short)0, acc, /*reuse_a=*/false, /*reuse_b=*/false);
  }

  const float bc = HAS_BIAS ? bias[col0 + mn] : 0.0f;
  float* __restrict__ o = out + (size_t)(row0 + 8 * g) * C + col0 + mn;
  #pragma unroll
  for (int r = 0; r < 8; ++r) {
    float v = acc[r] + bc;
    if (RELU_OUT) v = fmaxf(v, 0.f);
    o[(size_t)r * C] = v;
  }
}

// ---------------------------------------------------------------------------
// Self-loop + bias init:  agg[i][c] = b[c] + dinv[i]^2 * t[i][c]   (C = 64)
// One thread per (node, 4 channels).
// ---------------------------------------------------------------------------
__global__ void k_selfloop_bias(const float* __restrict__ t, const float* __restrict__ dinv,
                                const float* __restrict__ b, float* __restrict__ out, int n) {
  int tid = blockIdx.x * blockDim.x + threadIdx.x;
  if (tid >= n * 16) return;
  int i = tid >> 4;
  int c = (tid & 15) * 4;
  float di = dinv[i];
  float w  = di * di;
  const float4 tv = *(const float4*)(t + (size_t)i * 64 + c);
  const float4 bv = *(const float4*)(b + c);
  float4 o;
  o.x = bv.x + w * tv.x;
  o.y = bv.y + w * tv.y;
  o.z = bv.z + w * tv.z;
  o.w = bv.w + w * tv.w;
  *(float4*)(out + (size_t)i * 64 + c) = o;
}

// ---------------------------------------------------------------------------
// Edge scatter:  agg[dst][c] += enorm[e] * t[src][c]   (C = 64)
// Half-wave (16 lanes) per edge, float4 per lane -> b128 gather + 4 atomics.
// ---------------------------------------------------------------------------
__global__ void k_scatter(const float* __restrict__ t, const float* __restrict__ enorm,
                          const int* __restrict__ src, const int* __restrict__ dst,
                          float* __restrict__ out, int e) {
  int tid  = blockIdx.x * blockDim.x + threadIdx.x;
  int edge = tid >> 4;
  if (edge >= e) return;
  int c = (tid & 15) * 4;
  int s = src[edge];
  int d = dst[edge];
  float w = enorm[edge];
  const float4 v = *(const float4*)(t + (size_t)s * 64 + c);
  float* o = out + (size_t)d * 64 + c;
  unsafeAtomicAdd(o + 0, w * v.x);
  unsafeAtomicAdd(o + 1, w * v.y);
  unsafeAtomicAdd(o + 2, w * v.z);
  unsafeAtomicAdd(o + 3, w * v.w);
}

// ---------------------------------------------------------------------------
extern "C" void kernel_launch(void* const* d_in, const int* in_sizes, int n_in,
                              void* d_out, int out_size, void* d_ws, size_t ws_size,
                              hipStream_t stream) {
  const float* x   = (const float*)d_in[0];
  const int*   ei  = (const int*)d_in[1];
  const float* W1  = (const float*)d_in[2];
  const float* b1  = (const float*)d_in[3];
  const float* W2  = (const float*)d_in[4];
  const float* b2  = (const float*)d_in[5];
  const float* W3  = (const float*)d_in[6];
  const float* b3  = (const float*)d_in[7];
  const float* fw1 = (const float*)d_in[8];
  const float* fb1 = (const float*)d_in[9];
  const float* fw2 = (const float*)d_in[10];
  const float* fb2 = (const float*)d_in[11];
  float* out = (float*)d_out;

  const int N = GCN_N, E = GCN_E;
  const int* src = ei;        // edge_index[0]
  const int* dst = ei + E;    // edge_index[1]

  float* ws    = (float*)d_ws;
  float* dinv  = ws;                              // N
  float* enorm = dinv + N;                        // E
  float* bufT  = enorm + E;                       // N*64 (t = h@W)
  float* bufA  = bufT + (size_t)N * 64;           // N*64 (aggregated)
  float* bufH  = bufA + (size_t)N * 64;           // N*128 (fc1 output)

  // 1) symmetric GCN normalization
  k_init_deg  <<<(N + 255) / 256, 256, 0, stream>>>(dinv, N);
  k_accum_deg <<<(E + 255) / 256, 256, 0, stream>>>(dst, dinv, E);
  k_dinv      <<<(N + 255) / 256, 256, 0, stream>>>(dinv, N);
  k_edge_norm <<<(E + 255) / 256, 256, 0, stream>>>(src, dst, dinv, enorm, E);

  const dim3 blk(32, 4);
  const int  mt    = (N + 15) / 16;            // 6250 row tiles (exact)
  const int  nb_nc = (N * 16 + 255) / 256;     // node x 4-channel blocks
  const int  eb    = (E * 16 + 255) / 256;     // half-wave per edge

  // 2) GCN layer 1 (2->64, VALU)
  k_layer1<<<nb_nc, 256, 0, stream>>>(x, W1, bufT, N);
  k_selfloop_bias<<<nb_nc, 256, 0, stream>>>(bufT, dinv, b1, bufA, N);
  k_scatter<<<eb, 256, 0, stream>>>(bufT, enorm, src, dst, bufA, E);

  // 3) GCN layer 2 (relu of layer-1 folded into A load)
  k_gemm_wmma<64, 64, true, false, false><<<dim3(mt, 1), blk, 0, stream>>>(bufA, W2, nullptr, bufT);
  k_selfloop_bias<<<nb_nc, 256, 0, stream>>>(bufT, dinv, b2, bufA, N);
  k_scatter<<<eb, 256, 0, stream>>>(bufT, enorm, src, dst, bufA, E);

  // 4) GCN layer 3
  k_gemm_wmma<64, 64, true, false, false><<<dim3(mt, 1), blk, 0, stream>>>(bufA, W3, nullptr, bufT);
  k_selfloop_bias<<<nb_nc, 256, 0, stream>>>(bufT, dinv, b3, bufA, N);
  k_scatter<<<eb, 256, 0, stream>>>(bufT, enorm, src, dst, bufA, E);

  // 5) fc1: relu(agg3) @ fw1 + fb1, relu  -> N x 128
  k_gemm_wmma<64, 128, true, true, true><<<dim3(mt, 2), blk, 0, stream>>>(bufA, fw1, fb1, bufH);

  // 6) fc2: h128 @ fw2 + fb2 -> d_out (N x 32)
  k_gemm_wmma<128, 32, false, false, true><<<dim3(mt, 1), blk, 0, stream>>>(bufH, fw2, fb2, out);
}